// NCEAverageKinetics_67912022885096
// MI455X (gfx1250) — compile-verified
//
#include <hip/hip_runtime.h>
#include <hip/hip_bf16.h>

typedef __attribute__((ext_vector_type(2))) float v2f;
typedef __attribute__((ext_vector_type(8))) float v8f;

#define NCE_B     64
#define NCE_D     128
#define NCE_KP1   16385            // K+1
#define NCE_NDATA 240000
#define NCE_TILES 1025             // ceil(16385/16)
#define NCE_TPG   65               // tiles per block-group (16 groups * 65 >= 1025)
#define NCE_GRPX  16
#define NCE_NPART (NCE_GRPX * NCE_B)

// ---------------------------------------------------------------------------
// Kernel 1: gathered batched matvec via V_WMMA_F32_16X16X4_F32 + exp, partial sums
// ---------------------------------------------------------------------------
__global__ __launch_bounds__(256) void nce_logits_exp(
    const float* __restrict__ anchor,   // [64,128]
    const float* __restrict__ pair,     // [64,128]
    const float* __restrict__ bank,     // [240000,128]
    const int*   __restrict__ negs,     // [64,16385]
    float*       __restrict__ out_exp,  // [64,16385] (unnormalized exp)
    float*       __restrict__ partials) // [NCE_NPART]
{
    const int b    = blockIdx.y;
    const int lane = threadIdx.x & 31;
    const int wave = threadIdx.x >> 5;           // 8 waves / block
    const int half = lane >> 4;                  // 0 -> K{0,1}, 1 -> K{2,3}
    const int mrow = lane & 15;                  // A row within tile / C column N

    // Preload B fragments (anchor broadcast across 16 N columns), reused by all tiles.
    // Lane layout for B 4x16: lane holds anchor[4j + 2*half + {0,1}].
    const float* __restrict__ arow = anchor + b * NCE_D;
    v2f bf[32];
    #pragma unroll
    for (int j = 0; j < 32; ++j) {
        const float* p = arow + 4 * j + 2 * half;
        bf[j].x = p[0];
        bf[j].y = p[1];
    }

    const float invT = 1.0f / 0.07f;
    float lsum = 0.0f;

    const int t0 = blockIdx.x * NCE_TPG;
    int t1 = t0 + NCE_TPG;
    if (t1 > NCE_TILES) t1 = NCE_TILES;

    for (int t = t0 + wave; t < t1; t += 8) {
        // This lane's A row within the tile
        const int k  = t * 16 + mrow;
        const int kc = (k < NCE_KP1 - 1) ? k : (NCE_KP1 - 1);   // clamp tail
        const long idx = (long)negs[(long)b * NCE_KP1 + kc];
        const float* __restrict__ row =
            (kc == 0) ? (pair + (long)b * NCE_D) : (bank + idx * (long)NCE_D);

        v8f c = {0.f, 0.f, 0.f, 0.f, 0.f, 0.f, 0.f, 0.f};
        #pragma unroll
        for (int j = 0; j < 32; ++j) {
            v2f a;
            const float* ap = row + 4 * j + 2 * half;
            a.x = ap[0];
            a.y = ap[1];
            // D(16x16) += A(16x4) * B(4x16); every N column of C equals dot(row_M, anchor)
            c = __builtin_amdgcn_wmma_f32_16x16x4_f32(
                    false, a, false, bf[j], (short)0, c, false, false);
        }

        // Lanes with N==0 (lane 0 and lane 16) hold rows M = v + 8*half
        if (mrow == 0) {
            const int kbase = t * 16 + 8 * half;
            #pragma unroll
            for (int v = 0; v < 8; ++v) {
                const int kk = kbase + v;
                if (kk < NCE_KP1) {
                    const float e = __expf(c[v] * invT);
                    out_exp[(long)b * NCE_KP1 + kk] = e;
                    lsum += e;
                }
            }
        }
    }

    // Deterministic block reduction of lsum (only lanes 0/16 are nonzero per wave)
    __shared__ float wsum[8];
    #pragma unroll
    for (int off = 16; off > 0; off >>= 1)
        lsum += __shfl_down(lsum, off, 32);
    if (lane == 0) wsum[wave] = lsum;
    __syncthreads();
    if (threadIdx.x == 0) {
        float s = 0.0f;
        #pragma unroll
        for (int w = 0; w < 8; ++w) s += wsum[w];
        partials[blockIdx.y * gridDim.x + blockIdx.x] = s;
    }
}

// ---------------------------------------------------------------------------
// Kernel 2: fixed-order reduction of partials -> 1/Z
// ---------------------------------------------------------------------------
__global__ __launch_bounds__(256) void nce_reduce(
    const float* __restrict__ partials, int n, float* __restrict__ invZ)
{
    __shared__ float sm[256];
    float s = 0.0f;
    for (int i = threadIdx.x; i < n; i += 256) s += partials[i];
    sm[threadIdx.x] = s;
    __syncthreads();
    #pragma unroll
    for (int off = 128; off > 0; off >>= 1) {
        if (threadIdx.x < off) sm[threadIdx.x] += sm[threadIdx.x + off];
        __syncthreads();
    }
    if (threadIdx.x == 0) {
        const float total = sm[0];
        const float mean  = total / (float)(NCE_B * NCE_KP1);
        const float Z     = mean * (float)NCE_NDATA;
        *invZ = 1.0f / Z;
    }
}

// ---------------------------------------------------------------------------
// Kernel 3: scale out_c in place by 1/Z
// ---------------------------------------------------------------------------
__global__ __launch_bounds__(256) void nce_scale(
    float* __restrict__ out, int n, const float* __restrict__ invZ)
{
    const float s = *invZ;
    int i = blockIdx.x * blockDim.x + threadIdx.x;
    const int stride = gridDim.x * blockDim.x;
    for (; i < n; i += stride) out[i] *= s;
}

// ---------------------------------------------------------------------------
// Kernel 4: bulk copy of the memory bank (float4 vectorized)
// ---------------------------------------------------------------------------
__global__ __launch_bounds__(256) void nce_copy(
    const float4* __restrict__ src, float4* __restrict__ dst, long n4)
{
    long i = (long)blockIdx.x * blockDim.x + threadIdx.x;
    const long stride = (long)gridDim.x * blockDim.x;
    for (; i < n4; i += stride) dst[i] = src[i];
}

// ---------------------------------------------------------------------------
// Kernel 5: deterministic scatter of anchor rows into the copied bank
// (single block, ascending b -> duplicate indices: last b wins)
// ---------------------------------------------------------------------------
__global__ __launch_bounds__(128) void nce_scatter(
    const float* __restrict__ anchor, const int* __restrict__ idx,
    float* __restrict__ mem)
{
    const int tid = threadIdx.x;   // 128 threads, one per column
    for (int b = 0; b < NCE_B; ++b) {
        const long r = (long)idx[b];
        mem[r * NCE_D + tid] = anchor[(long)b * NCE_D + tid];
    }
}

// ---------------------------------------------------------------------------
extern "C" void kernel_launch(void* const* d_in, const int* in_sizes, int n_in,
                              void* d_out, int out_size, void* d_ws, size_t ws_size,
                              hipStream_t stream) {
    (void)in_sizes; (void)n_in; (void)out_size; (void)ws_size;

    const float* anchor = (const float*)d_in[0];   // [64,128]
    const float* pair   = (const float*)d_in[1];   // [64,128]
    const float* bank   = (const float*)d_in[2];   // [240000,128]
    const int*   negs   = (const int*)d_in[3];     // [64,16385]
    const int*   midx   = (const int*)d_in[4];     // [64]

    float* out      = (float*)d_out;
    float* out_c    = out;                                   // [64*16385]
    float* out_mem  = out + (long)NCE_B * NCE_KP1;           // [240000*128]

    float* partials = (float*)d_ws;                          // [1024]
    float* invZ     = partials + NCE_NPART;                  // [1]

    // 1) gathered matvec (WMMA f32) + exp + per-block partial sums
    nce_logits_exp<<<dim3(NCE_GRPX, NCE_B), 256, 0, stream>>>(
        anchor, pair, bank, negs, out_c, partials);

    // 2) reduce partials -> 1/Z
    nce_reduce<<<1, 256, 0, stream>>>(partials, NCE_NPART, invZ);

    // 3) normalize out_c in place
    nce_scale<<<1024, 256, 0, stream>>>(out_c, NCE_B * NCE_KP1, invZ);

    // 4) copy memory bank to output (30,720,000 floats = 7,680,000 float4)
    const long n4 = (long)NCE_NDATA * NCE_D / 4;
    nce_copy<<<3840, 256, 0, stream>>>(
        (const float4*)bank, (float4*)out_mem, n4);

    // 5) scatter anchor rows (deterministic ordering for duplicate indices)
    nce_scatter<<<1, 128, 0, stream>>>(anchor, midx, out_mem);
}